// AttentionMap_43782896615900
// MI455X (gfx1250) — compile-verified
//
#include <hip/hip_runtime.h>
#include <math.h>

// Problem constants (from reference): B=8, C=256, CR=32, H=W=48, N=2304
#define NB     8
#define NC     256
#define NCR    32
#define NPIX   2304            // 48*48
#define ROWS   32              // i-rows per workgroup (two 16-row WMMA subtiles)
#define SROW   2308            // padded LDS row stride (2308 % 64 == 4 -> staggered banks)
#define RCPOFF (ROWS * SROW)   // offset of per-row 1/sum array in LDS

typedef float v2f __attribute__((ext_vector_type(2)));
typedef float v8f __attribute__((ext_vector_type(8)));

// ---------------------------------------------------------------------------
// Projection: q[b,i,d] / k[b,d,i] for d in [0,32). One thread per pixel i,
// all 32 output channels in registers; weight reads are lane-uniform
// (scalar-cache broadcasts). Q stored [b][n][32] (A-fragment friendly),
// K stored [b][32][n] (B-fragment friendly).
// ---------------------------------------------------------------------------
__global__ __launch_bounds__(256) void proj_qk_kernel(
    const float* __restrict__ x,
    const float* __restrict__ w1, const float* __restrict__ b1,
    const float* __restrict__ w2, const float* __restrict__ b2,
    float* __restrict__ Q, float* __restrict__ Kt)
{
    const int b = blockIdx.y;
    const int i = blockIdx.x * 256 + threadIdx.x;
    const float* xb = x + (size_t)b * NC * NPIX + i;

    float qa[NCR], ka[NCR];
#pragma unroll
    for (int d = 0; d < NCR; ++d) { qa[d] = b1[d]; ka[d] = b2[d]; }

    for (int c = 0; c < NC; ++c) {
        const float xv = xb[(size_t)c * NPIX];
#pragma unroll
        for (int d = 0; d < NCR; ++d) {
            qa[d] = fmaf(w1[d * NC + c], xv, qa[d]);
            ka[d] = fmaf(w2[d * NC + c], xv, ka[d]);
        }
    }

    float* Qo = Q + ((size_t)b * NPIX + i) * NCR;
#pragma unroll
    for (int d = 0; d < NCR; ++d) Qo[d] = qa[d];
#pragma unroll
    for (int d = 0; d < NCR; ++d) Kt[((size_t)b * NCR + d) * NPIX + i] = ka[d];
}

// ---------------------------------------------------------------------------
// Projection: v[b,c,i] = w3[c,:] . x[b,:,i] + b3[c]. One thread per pixel,
// 64 output channels per thread (grid.y selects the 64-channel chunk).
// ---------------------------------------------------------------------------
__global__ __launch_bounds__(256) void proj_v_kernel(
    const float* __restrict__ x,
    const float* __restrict__ w3, const float* __restrict__ b3,
    float* __restrict__ V)
{
    const int b  = blockIdx.z;
    const int d0 = blockIdx.y * 64;
    const int i  = blockIdx.x * 256 + threadIdx.x;
    const float* xb = x + (size_t)b * NC * NPIX + i;

    float acc[64];
#pragma unroll
    for (int d = 0; d < 64; ++d) acc[d] = b3[d0 + d];

    for (int c = 0; c < NC; ++c) {
        const float xv = xb[(size_t)c * NPIX];
#pragma unroll
        for (int d = 0; d < 64; ++d)
            acc[d] = fmaf(w3[(d0 + d) * NC + c], xv, acc[d]);
    }
#pragma unroll
    for (int d = 0; d < 64; ++d)
        V[((size_t)b * NC + d0 + d) * NPIX + i] = acc[d];
}

// ---------------------------------------------------------------------------
// Fused attention: one workgroup per (batch, 32-row tile of N).
//   Phase 1: S[32 x 2304] = Qtile . K  via v_wmma_f32_16x16x4_f32 -> LDS
//            (two 16-row subtiles share each K B-fragment)
//   Softmax: row max, exp in place, keep 1/sum (folded into phase-2 output)
//   Phase 2: out[32c x 32i] per wave as a 2x2 grid of 16x16 WMMA tiles:
//            each V A-fragment and each P B-fragment feeds TWO of the FOUR
//            independent accumulator chains (halves both L2 and LDS traffic
//            per wmma, removes serial D->C dependencies).
// WMMA f32 fragment mapping (wave32):
//   A(16x4): lane l, vgpr v -> A[l&15][2*(l>>4)+v]   (float2 load)
//   B(4x16): lane l, vgpr v -> B[2*(l>>4)+v][l&15]
//   C/D    : vgpr r, lane l -> D[r + 8*(l>>4)][l&15]
// ---------------------------------------------------------------------------
__global__ __launch_bounds__(256) void attn_fused_kernel(
    const float* __restrict__ Q, const float* __restrict__ Kt,
    const float* __restrict__ V, const float* __restrict__ feat,
    float* __restrict__ out, const float* __restrict__ gamma_p)
{
    __shared__ float sS[ROWS * SROW + ROWS];   // 32-row score strip + 32 rcp sums

    const int wave = threadIdx.x >> 5;
    const int lane = threadIdx.x & 31;
    const int row  = lane & 15;            // M (phase1 i / phase2 c) or N col
    const int hi   = lane >> 4;            // upper half-wave selector

    const int b  = blockIdx.y;
    const int i0 = blockIdx.x * ROWS;

    // ---- preload Q A-fragments for the two 16-row subtiles (K=32 -> 8 steps)
    const float* Qb = Q + ((size_t)b * NPIX + i0) * NCR;
    v2f aq0[8], aq1[8];
#pragma unroll
    for (int kk = 0; kk < 8; ++kk) {
        aq0[kk] = *(const v2f*)(Qb + row * NCR + kk * 4 + 2 * hi);
        aq1[kk] = *(const v2f*)(Qb + (row + 16) * NCR + kk * 4 + 2 * hi);
    }

    // ---- phase 1: scores into LDS (K fragments shared by both subtiles) ----
    const float* Kb = Kt + (size_t)b * NCR * NPIX;
    for (int jt = wave; jt < NPIX / 16; jt += 8) {
        const int j0 = jt * 16;
        v8f acc0 = {}, acc1 = {};
#pragma unroll
        for (int kk = 0; kk < 8; ++kk) {
            const float* kp = Kb + (size_t)(kk * 4 + 2 * hi) * NPIX + j0 + row;
            v2f bf;
            bf.x = kp[0];
            bf.y = kp[NPIX];
            acc0 = __builtin_amdgcn_wmma_f32_16x16x4_f32(
                false, aq0[kk], false, bf, (short)0, acc0, false, false);
            acc1 = __builtin_amdgcn_wmma_f32_16x16x4_f32(
                false, aq1[kk], false, bf, (short)0, acc1, false, false);
        }
#pragma unroll
        for (int r = 0; r < 8; ++r) {
            sS[(r + 8 * hi) * SROW + j0 + row]        = acc0[r];
            sS[(16 + r + 8 * hi) * SROW + j0 + row]   = acc1[r];
        }
    }
    __syncthreads();

    // ---- softmax over each of the 32 rows (4 rows per wave) ----
#pragma unroll
    for (int rr = 0; rr < 4; ++rr) {
        const int r = wave * 4 + rr;
        float m = -INFINITY;
        for (int j = lane; j < NPIX; j += 32)
            m = fmaxf(m, sS[r * SROW + j]);
#pragma unroll
        for (int off = 16; off >= 1; off >>= 1)
            m = fmaxf(m, __shfl_xor(m, off, 32));
        float s = 0.0f;
        for (int j = lane; j < NPIX; j += 32) {
            const float e = __expf(sS[r * SROW + j] - m);
            sS[r * SROW + j] = e;
            s += e;
        }
#pragma unroll
        for (int off = 16; off >= 1; off >>= 1)
            s += __shfl_xor(s, off, 32);
        if (lane == 0) sS[RCPOFF + r] = 1.0f / s;
    }
    __syncthreads();

    // ---- phase 2: 2x2 WMMA tile grid per wave (c-tiles wave, wave+8) ----
    const float g    = gamma_p[0];
    const float rcp0 = sS[RCPOFF + row];        // output column i0 + row
    const float rcp1 = sS[RCPOFF + row + 16];   // output column i0 + 16 + row
    const float* Vb = V    + (size_t)b * NC * NPIX;
    const float* Fb = feat + (size_t)b * NC * NPIX;
    float*       Ob = out  + (size_t)b * NC * NPIX;

    const int c0 = wave * 16;                   // channel tile 0
    const int c1 = (wave + 8) * 16;             // channel tile 1
    v8f a00 = {}, a01 = {}, a10 = {}, a11 = {};
    const float* vrow0 = Vb + (size_t)(c0 + row) * NPIX + 2 * hi;
    const float* vrow1 = Vb + (size_t)(c1 + row) * NPIX + 2 * hi;
    const float* prow0 = sS + row * SROW + 2 * hi;
    const float* prow1 = sS + (row + 16) * SROW + 2 * hi;
    for (int j0 = 0; j0 < NPIX; j0 += 4) {
        const v2f af0 = *(const v2f*)(vrow0 + j0);  // V[c0+row][j0+2hi+{0,1}]
        const v2f af1 = *(const v2f*)(vrow1 + j0);  // V[c1+row][j0+2hi+{0,1}]
        const v2f bf0 = *(const v2f*)(prow0 + j0);  // P[row]   [j0+2hi+{0,1}]
        const v2f bf1 = *(const v2f*)(prow1 + j0);  // P[row+16][j0+2hi+{0,1}]
        a00 = __builtin_amdgcn_wmma_f32_16x16x4_f32(
            false, af0, false, bf0, (short)0, a00, false, false);
        a01 = __builtin_amdgcn_wmma_f32_16x16x4_f32(
            false, af0, false, bf1, (short)0, a01, false, false);
        a10 = __builtin_amdgcn_wmma_f32_16x16x4_f32(
            false, af1, false, bf0, (short)0, a10, false, false);
        a11 = __builtin_amdgcn_wmma_f32_16x16x4_f32(
            false, af1, false, bf1, (short)0, a11, false, false);
    }
#pragma unroll
    for (int r = 0; r < 8; ++r) {
        const int cc0 = c0 + r + 8 * hi;
        const int cc1 = c1 + r + 8 * hi;
        const size_t i00 = (size_t)cc0 * NPIX + i0 + row;
        const size_t i10 = (size_t)cc1 * NPIX + i0 + row;
        Ob[i00]      = fmaf(g * rcp0, a00[r], Fb[i00]);
        Ob[i00 + 16] = fmaf(g * rcp1, a01[r], Fb[i00 + 16]);
        Ob[i10]      = fmaf(g * rcp0, a10[r], Fb[i10]);
        Ob[i10 + 16] = fmaf(g * rcp1, a11[r], Fb[i10 + 16]);
    }
}

// ---------------------------------------------------------------------------
extern "C" void kernel_launch(void* const* d_in, const int* in_sizes, int n_in,
                              void* d_out, int out_size, void* d_ws, size_t ws_size,
                              hipStream_t stream)
{
    const float* feat  = (const float*)d_in[0];
    const float* w1    = (const float*)d_in[1];
    const float* b1    = (const float*)d_in[2];
    const float* w2    = (const float*)d_in[3];
    const float* b2    = (const float*)d_in[4];
    const float* w3    = (const float*)d_in[5];
    const float* b3    = (const float*)d_in[6];
    const float* gamma = (const float*)d_in[7];
    float* outp = (float*)d_out;

    // Workspace: Q [B,N,32] | Kt [B,32,N] | V [B,256,N]  (~23.6 MB fp32)
    float* Q  = (float*)d_ws;
    float* Kt = Q  + (size_t)NB * NPIX * NCR;
    float* V  = Kt + (size_t)NB * NCR * NPIX;

    proj_qk_kernel<<<dim3(NPIX / 256, NB), 256, 0, stream>>>(feat, w1, b1, w2, b2, Q, Kt);
    proj_v_kernel <<<dim3(NPIX / 256, NC / 64, NB), 256, 0, stream>>>(feat, w3, b3, V);
    attn_fused_kernel<<<dim3(NPIX / ROWS, NB), 256, 0, stream>>>(Q, Kt, V, feat, outp, gamma);
}